// NodeModel_6244882448874
// MI455X (gfx1250) — compile-verified
//
#include <hip/hip_runtime.h>

typedef __attribute__((ext_vector_type(2))) float v2f;
typedef __attribute__((ext_vector_type(8))) float v8f;

#define N_NODES 50000
#define N_EDGES 800000
#define K_DIM   256      // MLP input/hidden width
#define NEG_SLOPE 0.01f

// ---------------------------------------------------------------------------
// Kernel 1: scatter-add edge_attr[e, f] into agg[col[e], f]
// 51.2M threads, one f32 atomic each; agg (12.8MB) stays L2-resident.
// ---------------------------------------------------------------------------
__global__ void scatter_add_kernel(const int* __restrict__ edge_index, // [2, E]
                                   const float* __restrict__ edge_attr, // [E, 64]
                                   float* __restrict__ agg,             // [N, 64]
                                   int total)
{
    int i = blockIdx.x * blockDim.x + threadIdx.x;
    if (i >= total) return;
    int e = i >> 6;        // edge id
    int f = i & 63;        // feature id
    int dst = edge_index[N_EDGES + e];   // col = edge_index[1]
    atomicAdd(&agg[dst * 64 + f], edge_attr[i]);
}

// ---------------------------------------------------------------------------
// Kernel 2: H = concat([x, agg, u[batch]], axis=1) -> [N, 256], float4 lanes
// ---------------------------------------------------------------------------
__global__ void concat_kernel(const float* __restrict__ x,     // [N,128]
                              const float* __restrict__ agg,   // [N,64]
                              const float* __restrict__ u,     // [G,64]
                              const int*   __restrict__ batch, // [N]
                              float* __restrict__ H,           // [N,256]
                              int total4)                      // N*64
{
    int i = blockIdx.x * blockDim.x + threadIdx.x;
    if (i >= total4) return;
    int row = i >> 6;
    int c   = (i & 63) << 2;   // column of this float4
    float4 v;
    if (c < 128)      v = *(const float4*)&x[row * 128 + c];
    else if (c < 192) v = *(const float4*)&agg[row * 64 + (c - 128)];
    else              v = *(const float4*)&u[batch[row] * 64 + (c - 192)];
    *(float4*)&H[row * 256 + c] = v;
}

// ---------------------------------------------------------------------------
// Kernel 3: Y = act(A @ W + bias) via V_WMMA_F32_16X16X4_F32
//   A: [M, 256]  W: [256, NOUT]  Y: [M, NOUT], M % 16 == 0
// Block = 256 threads = 8 waves; each wave owns one 16-row strip x full NOUT.
// W is staged through LDS in 32-row K chunks using a *paired-K* layout:
//   buf[p*NOUT + n] = { W[2p][n], W[2p+1][n] }  (v2f, 8B aligned)
// so every WMMA B operand is one ds_load_b64 into an even VGPR pair (no
// repacking movs). Column index is XOR-swizzled with (p&1)<<4 so the two
// lane-halves of a wave hit disjoint 32-bank groups (conflict-free b64).
// Chunks are double-buffered: two __syncthreads per chunk; the global fetch
// of chunk c+1 overlaps WMMA compute of chunk c.
// Safe for in-place Y == A (each row is read only by the wave that writes it).
// ---------------------------------------------------------------------------
template <int NOUT, bool ACT>
__global__ void __launch_bounds__(256)
gemm_wmma_kernel(const float* __restrict__ A,
                 const float* __restrict__ W,
                 const float* __restrict__ bias,
                 float* __restrict__ Y,
                 int M)
{
    constexpr int KC     = 32;             // K rows per chunk
    constexpr int NP     = KC / 2;         // 16 k-pairs per chunk
    constexpr int NT     = NOUT / 16;      // 16-col tiles
    constexpr int NCHUNK = K_DIM / KC;     // 8
    constexpr int NHALF  = NOUT / 2;       // float4 elements per k-pair row
    constexpr int PER_THREAD = (NP * NHALF) / 256;  // 8 (NOUT=256) / 4 (128)

    __shared__ v2f sW[2][NP * NOUT];       // 64KB / 32KB, double buffered

    const int tid  = threadIdx.x;
    const int wave = tid >> 5;
    const int lane = tid & 31;
    const int m    = lane & 15;            // M (and N) index within tile
    const int hi   = lane >> 4;            // K-pair parity / C-row-half select
    const int row0 = (blockIdx.x * 8 + wave) * 16;
    const bool active = (row0 < M);

    // accumulators: all 8 slots of a lane share column n = nt*16 + m
    v8f acc[NT];
    #pragma unroll
    for (int nt = 0; nt < NT; ++nt) {
        float b0 = bias[nt * 16 + m];
        #pragma unroll
        for (int j = 0; j < 8; ++j) acc[nt][j] = b0;
    }

    const float* arow = A + (size_t)(row0 + m) * K_DIM + 2 * hi;

    // ---- W chunk fetch (global -> regs), paired-K interleave ----
    float4 stage[PER_THREAD];
    auto fetch = [&](int c) {
        #pragma unroll
        for (int i = 0; i < PER_THREAD; ++i) {
            int idx = tid + i * 256;
            int p   = idx / NHALF;          // k-pair within chunk
            int n   = (idx % NHALF) * 2;    // column pair
            const float* g = W + (size_t)(c * KC + 2 * p) * NOUT + n;
            float2 r0 = *(const float2*)g;
            float2 r1 = *(const float2*)(g + NOUT);
            stage[i] = make_float4(r0.x, r1.x, r0.y, r1.y);
        }
    };

    fetch(0);
    for (int c = 0; c < NCHUNK; ++c) {
        v2f* buf = sW[c & 1];
        // commit staged chunk c (swizzled column)
        #pragma unroll
        for (int i = 0; i < PER_THREAD; ++i) {
            int idx = tid + i * 256;
            int p   = idx / NHALF;
            int n   = (idx % NHALF) * 2;
            int nsw = n ^ ((p & 1) << 4);
            *(float4*)&buf[p * NOUT + nsw] = stage[i];
        }
        __syncthreads();
        if (c + 1 < NCHUNK) fetch(c + 1);   // overlaps compute below

        if (active) {
            // A chunk in native WMMA 16x4 layout: lane holds A[row0+m][4t+2hi .. +1]
            v2f areg[KC / 4];
            #pragma unroll
            for (int t = 0; t < KC / 4; ++t)
                areg[t] = *(const v2f*)(arow + c * KC + 4 * t);

            #pragma unroll
            for (int nt = 0; nt < NT; ++nt) {
                const int n   = nt * 16 + m;
                const int nsw = n ^ (hi << 4);
                #pragma unroll
                for (int t = 0; t < KC / 4; ++t) {
                    const int p = 2 * t + hi;
                    v2f b = buf[p * NOUT + nsw];
                    acc[nt] = __builtin_amdgcn_wmma_f32_16x16x4_f32(
                        false, areg[t], false, b, (short)0, acc[nt],
                        false, false);
                }
            }
        }
        __syncthreads();   // buffer c&1 free for reuse at c+2
    }

    if (active) {
        // C/D layout: VGPR j -> row = row0 + hi*8 + j, col = nt*16 + m
        #pragma unroll
        for (int nt = 0; nt < NT; ++nt) {
            const int n = nt * 16 + m;
            #pragma unroll
            for (int j = 0; j < 8; ++j) {
                float v = acc[nt][j];
                if (ACT) v = (v >= 0.0f) ? v : NEG_SLOPE * v;
                Y[(size_t)(row0 + hi * 8 + j) * NOUT + n] = v;
            }
        }
    }
}

// ---------------------------------------------------------------------------
// Host launcher
// Inputs: 0:x 1:edge_index 2:edge_attr 3:u 4:batch 5:w1 6:b1 7:w2 8:b2 9:w3 10:b3
// Workspace: agg [50000*64] f32, then H [50000*256] f32  (64 MB total)
// ---------------------------------------------------------------------------
extern "C" void kernel_launch(void* const* d_in, const int* in_sizes, int n_in,
                              void* d_out, int out_size, void* d_ws, size_t ws_size,
                              hipStream_t stream)
{
    const float* x          = (const float*)d_in[0];
    const int*   edge_index = (const int*)d_in[1];
    const float* edge_attr  = (const float*)d_in[2];
    const float* u          = (const float*)d_in[3];
    const int*   batch      = (const int*)d_in[4];
    const float* w1         = (const float*)d_in[5];
    const float* b1         = (const float*)d_in[6];
    const float* w2         = (const float*)d_in[7];
    const float* b2         = (const float*)d_in[8];
    const float* w3         = (const float*)d_in[9];
    const float* b3         = (const float*)d_in[10];
    float*       out        = (float*)d_out;

    float* agg = (float*)d_ws;                 // [50000, 64]
    float* H   = agg + (size_t)N_NODES * 64;   // [50000, 256]

    // zero the scatter accumulator every call (graph-capture safe)
    hipMemsetAsync(agg, 0, (size_t)N_NODES * 64 * sizeof(float), stream);

    // 1) scatter-add edge features
    {
        int total  = N_EDGES * 64;
        int blocks = (total + 255) / 256;
        scatter_add_kernel<<<blocks, 256, 0, stream>>>(edge_index, edge_attr, agg, total);
    }

    // 2) build H = [x | agg | u[batch]]  (float4 granularity)
    {
        int total4 = N_NODES * 64;
        int blocks = (total4 + 255) / 256;
        concat_kernel<<<blocks, 256, 0, stream>>>(x, agg, u, batch, H, total4);
    }

    // 3) three WMMA GEMMs; layers 1-2 run in place on H
    {
        int strips = N_NODES / 16;             // 3125 (50000 % 16 == 0)
        int blocks = (strips + 7) / 8;         // 391
        gemm_wmma_kernel<256, true ><<<blocks, 256, 0, stream>>>(H, w1, b1, H,   N_NODES);
        gemm_wmma_kernel<256, true ><<<blocks, 256, 0, stream>>>(H, w2, b2, H,   N_NODES);
        gemm_wmma_kernel<128, false><<<blocks, 256, 0, stream>>>(H, w3, b3, out, N_NODES);
    }
}